// ForwardPredictionHead_84791244357848
// MI455X (gfx1250) — compile-verified
//
#include <hip/hip_runtime.h>
#include <hip/hip_bf16.h>
#include <math.h>

typedef __attribute__((ext_vector_type(2))) float v2f;
typedef __attribute__((ext_vector_type(8))) float v8f;

#define HWD   65536   // 256*256
#define HWSH  16
#define NCLS  12
#define NPAR  25

// ---------------------------------------------------------------------------
// K1: GroupNorm statistics per (b, g). One block per (b*groups + g).
// Group data is channel-contiguous, so each block reduces a contiguous chunk.
// Double accumulators: group sizes up to 262144 f32 values would lose ~1e-2
// relative accuracy in f32; f64 keeps it exact enough.
// ---------------------------------------------------------------------------
__global__ __launch_bounds__(256)
void gn_stats_kernel(const float* __restrict__ x, int C, int HW, int groups,
                     float* __restrict__ mu, float* __restrict__ rstd)
{
    const int bg = blockIdx.x;
    const int cg = C / groups;
    const long long n = (long long)cg * HW;
    const float* base = x + (long long)bg * n;

    double s = 0.0, s2 = 0.0;
    for (long long i = threadIdx.x; i < n; i += blockDim.x) {
        float v = base[i];
        s += (double)v;
        s2 += (double)v * (double)v;
    }
    __shared__ double sh_s[256];
    __shared__ double sh_q[256];
    sh_s[threadIdx.x] = s;
    sh_q[threadIdx.x] = s2;
    __syncthreads();
    for (int off = 128; off > 0; off >>= 1) {
        if ((int)threadIdx.x < off) {
            sh_s[threadIdx.x] += sh_s[threadIdx.x + off];
            sh_q[threadIdx.x] += sh_q[threadIdx.x + off];
        }
        __syncthreads();
    }
    if (threadIdx.x == 0) {
        double mean = sh_s[0] / (double)n;
        double var  = sh_q[0] / (double)n - mean * mean;
        mu[bg]   = (float)mean;
        rstd[bg] = (float)(1.0 / sqrt(var + 1e-5));
    }
}

// ---------------------------------------------------------------------------
// K2: pooled[b][c] = mean_{hw} relu( gn(x) )  — one block per (b, c)
// ---------------------------------------------------------------------------
__global__ __launch_bounds__(256)
void pooled_kernel(const float* __restrict__ x, int C, int HW, int groups,
                   const float* __restrict__ mu, const float* __restrict__ rstd,
                   const float* __restrict__ w, const float* __restrict__ bia,
                   float* __restrict__ pooled)
{
    const int bc = blockIdx.x;
    const int c  = bc % C;
    const int b  = bc / C;
    const int g  = c / (C / groups);
    const float m  = mu[b * groups + g];
    const float rs = rstd[b * groups + g];
    const float scale = rs * w[c];
    const float shift = bia[c] - m * scale;
    const float* base = x + (long long)bc * HW;

    float s = 0.f;
    for (int i = threadIdx.x; i < HW; i += blockDim.x)
        s += fmaxf(fmaf(base[i], scale, shift), 0.f);

    __shared__ float sh[256];
    sh[threadIdx.x] = s;
    __syncthreads();
    for (int off = 128; off > 0; off >>= 1) {
        if ((int)threadIdx.x < off) sh[threadIdx.x] += sh[threadIdx.x + off];
        __syncthreads();
    }
    if (threadIdx.x == 0) pooled[bc] = sh[0] / (float)HW;
}

// ---------------------------------------------------------------------------
// K3: feat[b][o] = (g64 + g256 + g512)/3  (tiny GEMVs, 8 blocks x 256 thr)
// ---------------------------------------------------------------------------
__global__ __launch_bounds__(256)
void feat_kernel(const float* __restrict__ p64, const float* __restrict__ p256,
                 const float* __restrict__ p512,
                 const float* __restrict__ w64,  const float* __restrict__ b64,
                 const float* __restrict__ w256, const float* __restrict__ b256,
                 const float* __restrict__ w512, const float* __restrict__ b512,
                 float* __restrict__ feat)
{
    const int b = blockIdx.x;
    const int o = threadIdx.x;          // 0..255
    float s = b64[o] + b256[o] + b512[o];
    for (int c = 0; c < 64;  ++c) s = fmaf(p64 [b*64  + c], w64 [o*64  + c], s);
    for (int c = 0; c < 256; ++c) s = fmaf(p256[b*256 + c], w256[o*256 + c], s);
    for (int c = 0; c < 512; ++c) s = fmaf(p512[b*512 + c], w512[o*512 + c], s);
    feat[b*256 + o] = s * (1.f / 3.f);
}

// ---------------------------------------------------------------------------
// K4: params[b][n][p] = feat[b]·wctrl[p,:256] + clip[n]·wctrl[p,256:] + bctrl[p]
// ---------------------------------------------------------------------------
__global__ __launch_bounds__(32)
void params_kernel(const float* __restrict__ feat, const float* __restrict__ clip,
                   const float* __restrict__ wctrl, const float* __restrict__ bctrl,
                   float* __restrict__ params)
{
    const int bn = blockIdx.x;          // b*12 + n
    const int b  = bn / NCLS;
    const int n  = bn % NCLS;
    const int p  = threadIdx.x;
    if (p < NPAR) {
        float s = bctrl[p];
        const float* wrow = wctrl + p * 512;
        const float* fb   = feat + b * 256;
        const float* cn   = clip + n * 256;
        for (int k = 0; k < 256; ++k) s = fmaf(fb[k], wrow[k], s);
        for (int k = 0; k < 256; ++k) s = fmaf(cn[k], wrow[256 + k], s);
        params[bn * NPAR + p] = s;
    }
}

// ---------------------------------------------------------------------------
// K5: fused decoder head. Per wave: two 16-pixel tiles (32 contiguous pixels).
//   h  = relu(gn(decoder))                     (VALU, fused into loads)
//   x  = Wpre(4x16) @ h(16x16) + bpre          (4 accumulating WMMA 16x16x4)
//   per class n: y = relu(W1(4x4) @ x + b1)    (1 WMMA 16x16x4 per tile)
//                out = w2·y + b2               (VALU dot4, full-wave store)
// A operands (Wpre chunks, W1) are zero-padded to 16 rows; pad rows of D are
// never read. EXEC stays all-ones (selects only, no divergence).
// ---------------------------------------------------------------------------
__global__ __launch_bounds__(256)
void fused_head_kernel(const float* __restrict__ dec,     // (8,16,256,256)
                       const float* __restrict__ gn16w,
                       const float* __restrict__ gn16b,
                       const float* __restrict__ wpre,    // (4,16)
                       const float* __restrict__ bpre,    // (4,)
                       const float* __restrict__ params,  // (8,12,25)
                       const float* __restrict__ dmu,     // (8,4)
                       const float* __restrict__ drstd,   // (8,4)
                       float* __restrict__ out)           // (8,12,256,256)
{
    const int b      = blockIdx.y;
    const int lane   = threadIdx.x & 31;
    const int wave   = threadIdx.x >> 5;
    const int wPerB  = gridDim.x * (blockDim.x >> 5);
    const int wInB   = blockIdx.x * (blockDim.x >> 5) + wave;

    const int m  = lane & 15;   // row (A) / column (B,D) index inside tile
    const int hi = lane >> 4;   // half-wave: selects K=0,1 vs K=2,3 rows

    // Per-lane GN scale/shift for its 8 channels: chan(k,j) = 4k + 2*hi + j.
    // chunk k's 4 channels are exactly group k (C=16, groups=4).
    float sc[8], sh[8];
#pragma unroll
    for (int k = 0; k < 4; ++k) {
        const float muk = dmu[b * 4 + k];
        const float rsk = drstd[b * 4 + k];
#pragma unroll
        for (int j = 0; j < 2; ++j) {
            const int c = 4 * k + 2 * hi + j;
            const float wv = gn16w[c];
            sc[k * 2 + j] = rsk * wv;
            sh[k * 2 + j] = gn16b[c] - muk * rsk * wv;
        }
    }

    // A operands for the 16->4 projection: Wpre column-chunks, rows 4..15 = 0.
    v2f aWp[4];
#pragma unroll
    for (int k = 0; k < 4; ++k) {
        const int c0 = 4 * k + 2 * hi;
        v2f a; a.x = 0.f; a.y = 0.f;
        if (m < 4) { a.x = wpre[m * 16 + c0]; a.y = wpre[m * 16 + c0 + 1]; }
        aWp[k] = a;
    }

    // A operands for the per-class 4x4 matmuls (dynamic W1 from params).
    const float* Pb = params + b * NCLS * NPAR;
    v2f aW1[NCLS];
#pragma unroll
    for (int n = 0; n < NCLS; ++n) {
        v2f a; a.x = 0.f; a.y = 0.f;
        if (m < 4) {
            a.x = Pb[n * NPAR + m * 4 + 2 * hi];
            a.y = Pb[n * NPAR + m * 4 + 2 * hi + 1];
        }
        aW1[n] = a;
    }

    // C operand carrying b_precls on rows 0..3 (rows 4..15 and hi-half zero).
    v8f cPre = {};
#pragma unroll
    for (int r = 0; r < 4; ++r)
        cPre[r] = (lane < 16) ? bpre[r] : 0.f;

    const v8f zeroC = {};
    const float* decb = dec + (size_t)b * 16 * HWD;
    float* outb       = out + (size_t)b * NCLS * HWD;

    for (int tp = wInB; tp < (HWD / 32); tp += wPerB) {
        const int P = tp * 32;   // tileA pixels [P,P+16), tileB [P+16,P+32)

        // ---- load + GN + relu straight into WMMA B-operand layout ----
        v2f bhA[4], bhB[4];
#pragma unroll
        for (int k = 0; k < 4; ++k) {
            const int c0 = 4 * k + 2 * hi;
            const float* p0 = decb + ((size_t)c0       << HWSH) + P + m;
            const float* p1 = decb + ((size_t)(c0 + 1) << HWSH) + P + m;
            const float a0 = p0[0],  a1 = p1[0];    // tile A
            const float q0 = p0[16], q1 = p1[16];   // tile B
            bhA[k].x = fmaxf(fmaf(a0, sc[2*k],   sh[2*k]),   0.f);
            bhA[k].y = fmaxf(fmaf(a1, sc[2*k+1], sh[2*k+1]), 0.f);
            bhB[k].x = fmaxf(fmaf(q0, sc[2*k],   sh[2*k]),   0.f);
            bhB[k].y = fmaxf(fmaf(q1, sc[2*k+1], sh[2*k+1]), 0.f);
        }

        // ---- x = Wpre @ h + bpre : 4 accumulating WMMAs per tile ----
        v8f xA = cPre, xB = cPre;
#pragma unroll
        for (int k = 0; k < 4; ++k) {
            xA = __builtin_amdgcn_wmma_f32_16x16x4_f32(
                     false, aWp[k], false, bhA[k], (short)0, xA, false, false);
            xB = __builtin_amdgcn_wmma_f32_16x16x4_f32(
                     false, aWp[k], false, bhB[k], (short)0, xB, false, false);
        }

        // ---- convert x (D layout, rows 0..3) into B-operand layout ----
        v2f bxA, bxB;
        {
            const float t2 = __shfl(xA[2], m, 32);
            const float t3 = __shfl(xA[3], m, 32);
            bxA.x = (lane < 16) ? xA[0] : t2;
            bxA.y = (lane < 16) ? xA[1] : t3;
            const float u2 = __shfl(xB[2], m, 32);
            const float u3 = __shfl(xB[3], m, 32);
            bxB.x = (lane < 16) ? xB[0] : u2;
            bxB.y = (lane < 16) ? xB[1] : u3;
        }

        // ---- per class: WMMA 4x4 matmul, bias+relu, dot4, packed store ----
#pragma unroll
        for (int n = 0; n < NCLS; ++n) {
            const float* pc = Pb + n * NPAR;     // uniform -> scalar loads
            const float w20 = pc[16], w21 = pc[17], w22 = pc[18], w23 = pc[19];
            const float b10 = pc[20], b11 = pc[21], b12 = pc[22], b13 = pc[23];
            const float b2  = pc[24];

            v8f yA = __builtin_amdgcn_wmma_f32_16x16x4_f32(
                         false, aW1[n], false, bxA, (short)0, zeroC, false, false);
            v8f yB = __builtin_amdgcn_wmma_f32_16x16x4_f32(
                         false, aW1[n], false, bxB, (short)0, zeroC, false, false);

            float oA = b2;
            oA = fmaf(fmaxf(yA[0] + b10, 0.f), w20, oA);
            oA = fmaf(fmaxf(yA[1] + b11, 0.f), w21, oA);
            oA = fmaf(fmaxf(yA[2] + b12, 0.f), w22, oA);
            oA = fmaf(fmaxf(yA[3] + b13, 0.f), w23, oA);

            float oB = b2;
            oB = fmaf(fmaxf(yB[0] + b10, 0.f), w20, oB);
            oB = fmaf(fmaxf(yB[1] + b11, 0.f), w21, oB);
            oB = fmaf(fmaxf(yB[2] + b12, 0.f), w22, oB);
            oB = fmaf(fmaxf(yB[3] + b13, 0.f), w23, oB);

            // pack tileB results into lanes 16..31 -> one full 128B store
            const float tB  = __shfl(oB, m, 32);
            const float val = (lane < 16) ? oA : tB;
            outb[((size_t)n << HWSH) + P + lane] = val;
        }
    }
}

// ---------------------------------------------------------------------------
extern "C" void kernel_launch(void* const* d_in, const int* in_sizes, int n_in,
                              void* d_out, int out_size, void* d_ws, size_t ws_size,
                              hipStream_t stream)
{
    const float* feat512 = (const float*)d_in[0];
    const float* feat256 = (const float*)d_in[1];
    const float* feat64  = (const float*)d_in[2];
    const float* clip    = (const float*)d_in[3];
    const float* dec     = (const float*)d_in[4];
    const float* gn64w   = (const float*)d_in[5];
    const float* gn64b   = (const float*)d_in[6];
    const float* gn256w  = (const float*)d_in[7];
    const float* gn256b  = (const float*)d_in[8];
    const float* gn512w  = (const float*)d_in[9];
    const float* gn512b  = (const float*)d_in[10];
    const float* gn16w   = (const float*)d_in[11];
    const float* gn16b   = (const float*)d_in[12];
    const float* wg64    = (const float*)d_in[13];
    const float* bg64    = (const float*)d_in[14];
    const float* wg256   = (const float*)d_in[15];
    const float* bg256   = (const float*)d_in[16];
    const float* wg512   = (const float*)d_in[17];
    const float* bg512   = (const float*)d_in[18];
    const float* wpre    = (const float*)d_in[19];
    const float* bpre    = (const float*)d_in[20];
    const float* wctrl   = (const float*)d_in[21];
    const float* bctrl   = (const float*)d_in[22];
    float* out = (float*)d_out;

    // workspace layout (floats)
    float* ws = (float*)d_ws;
    float* mu64    = ws + 0;      // 32
    float* rstd64  = ws + 32;     // 32
    float* mu256   = ws + 64;
    float* rstd256 = ws + 96;
    float* mu512   = ws + 128;
    float* rstd512 = ws + 160;
    float* dmu     = ws + 192;    // 32
    float* drstd   = ws + 224;    // 32
    float* p64     = ws + 256;    // 8*64
    float* p256    = ws + 768;    // 8*256
    float* p512    = ws + 2816;   // 8*512
    float* feat    = ws + 6912;   // 8*256
    float* params  = ws + 8960;   // 8*12*25
    // total 11360 floats (~45 KB)

    // GroupNorm statistics (B*groups = 32 blocks each)
    gn_stats_kernel<<<32, 256, 0, stream>>>(feat64,  64,  64*64, 4, mu64,  rstd64);
    gn_stats_kernel<<<32, 256, 0, stream>>>(feat256, 256, 32*32, 4, mu256, rstd256);
    gn_stats_kernel<<<32, 256, 0, stream>>>(feat512, 512, 16*16, 4, mu512, rstd512);
    gn_stats_kernel<<<32, 256, 0, stream>>>(dec,     16,  HWD,   4, dmu,   drstd);

    // GAP branches -> pooled per channel
    pooled_kernel<<<8*64,  256, 0, stream>>>(feat64,  64,  64*64, 4, mu64,  rstd64,  gn64w,  gn64b,  p64);
    pooled_kernel<<<8*256, 256, 0, stream>>>(feat256, 256, 32*32, 4, mu256, rstd256, gn256w, gn256b, p256);
    pooled_kernel<<<8*512, 256, 0, stream>>>(feat512, 512, 16*16, 4, mu512, rstd512, gn512w, gn512b, p512);

    // feat (8,256) and dynamic params (8,12,25)
    feat_kernel<<<8, 256, 0, stream>>>(p64, p256, p512,
                                       wg64, bg64, wg256, bg256, wg512, bg512, feat);
    params_kernel<<<8*NCLS, 32, 0, stream>>>(feat, clip, wctrl, bctrl, params);

    // fused decoder head (WMMA): 64 blocks x 8 waves per batch, 4 tile-pairs/wave
    fused_head_kernel<<<dim3(64, 8), 256, 0, stream>>>(
        dec, gn16w, gn16b, wpre, bpre, params, dmu, drstd, out);
}